// SemiSupervisedGraphAttentionLayer_43499428774651
// MI455X (gfx1250) — compile-verified
//
#include <hip/hip_runtime.h>
#include <hip/hip_bf16.h>

#define DIM 128
#define HEADS 4
#define HDIM 32

typedef __attribute__((ext_vector_type(2))) float v2f;
typedef __attribute__((ext_vector_type(8))) float v8f;

// ---------------------------------------------------------------------------
// K1: h = X @ W^T + b   via V_WMMA_F32_16X16X4_F32.
// One block = 8 waves = one 16-row strip of h; wave w owns columns [16w,16w+16).
// ---------------------------------------------------------------------------
__global__ __launch_bounds__(256) void gat_h_gemm(
    const float* __restrict__ X, const float* __restrict__ W,
    const float* __restrict__ bl, float* __restrict__ H, int nn)
{
  __shared__ float lds_a[16 * 132];          // padded stride: bank-conflict free
  const int m0  = blockIdx.x * 16;
  const int tid = threadIdx.x;

  // Cooperative coalesced load of the 16x128 A tile (2 x float4 per thread).
  {
    int r = tid >> 4;                        // 0..15
    int c = (tid & 15) * 8;                  // 0..120
    int row = m0 + r; if (row >= nn) row = nn - 1;
    const float4* src = (const float4*)(X + (size_t)row * DIM + c);
    float4 p0 = src[0], p1 = src[1];
    float* dst = &lds_a[r * 132 + c];
    dst[0]=p0.x; dst[1]=p0.y; dst[2]=p0.z; dst[3]=p0.w;
    dst[4]=p1.x; dst[5]=p1.y; dst[6]=p1.z; dst[7]=p1.w;
  }
  __syncthreads();

  const int wave    = tid >> 5;
  const int lane    = tid & 31;
  const int j0      = wave * 16;
  const int halfsel = lane >> 4;             // 0 | 1 selects K pair
  const int l15     = lane & 15;

  v8f acc = {};
  const float* arow = &lds_a[l15 * 132];                // A row m = l15
  const float* wrow = &W[(size_t)(j0 + l15) * DIM];     // B col n = l15 (= W row)

  #pragma unroll
  for (int k0 = 0; k0 < DIM; k0 += 4) {
    const int ka = k0 + halfsel * 2;
    v2f a;  a.x  = arow[ka]; a.y  = arow[ka + 1];       // ds_load_b64
    v2f bb; bb.x = wrow[ka]; bb.y = wrow[ka + 1];       // global_load_b64
    acc = __builtin_amdgcn_wmma_f32_16x16x4_f32(
        false, a, false, bb, (short)0, acc, false, false);
  }

  // C/D layout: vgpr r -> (M = r + 8*halfsel, N = l15)
  const int col  = j0 + l15;
  const float bv = bl[col];
  const int rbase = m0 + halfsel * 8;
  #pragma unroll
  for (int r = 0; r < 8; ++r) {
    int row = rbase + r;
    if (row < nn) H[(size_t)row * DIM + col] = acc[r] + bv;
  }
}

// ---------------------------------------------------------------------------
// K2: per-node attention projections.  One wave per node:
//   es[n,h] = exp(h[n]·Wa[h, :128]),  gd[n,h] = exp(h[n]·Wa[h,128:] + b[h])
// ---------------------------------------------------------------------------
__global__ __launch_bounds__(256) void gat_att_proj(
    const float* __restrict__ H, const float* __restrict__ Wa,
    const float* __restrict__ ba, float* __restrict__ es,
    float* __restrict__ gd, int nn)
{
  const int lane = threadIdx.x & 31;
  const int node = (blockIdx.x * blockDim.x + threadIdx.x) >> 5;
  if (node >= nn) return;

  const float* hrow = &H[(size_t)node * DIM];
  float hv[4];
  #pragma unroll
  for (int i = 0; i < 4; ++i) hv[i] = hrow[lane + 32 * i];

  float ls[HEADS] = {0,0,0,0}, ld[HEADS] = {0,0,0,0};
  #pragma unroll
  for (int h = 0; h < HEADS; ++h) {
    const float* wr = &Wa[h * 2 * DIM];
    #pragma unroll
    for (int i = 0; i < 4; ++i) {
      const int j = lane + 32 * i;
      ls[h] += hv[i] * wr[j];
      ld[h] += hv[i] * wr[DIM + j];
    }
  }
  #pragma unroll
  for (int h = 0; h < HEADS; ++h) {
    #pragma unroll
    for (int off = 16; off > 0; off >>= 1) {
      ls[h] += __shfl_xor(ls[h], off, 32);
      ld[h] += __shfl_xor(ld[h], off, 32);
    }
  }
  if (lane == 0) {
    #pragma unroll
    for (int h = 0; h < HEADS; ++h) {
      es[node * HEADS + h] = __expf(ls[h]);
      gd[node * HEADS + h] = __expf(ld[h] + ba[h]);
    }
  }
}

// ---------------------------------------------------------------------------
// K3: edge pass.  One wave per edge (grid-stride); lane = head-dim index.
//   S[dst, h*32+d] += es[src,h] * h[src,d]   (global f32 atomics, L2-resident)
//   Z[h]          += es[src,h] * gd[dst,h]   (LDS-reduced, 1 atomic/block/head)
// ---------------------------------------------------------------------------
__global__ __launch_bounds__(256) void gat_edge_scatter(
    const int* __restrict__ eidx, const float* __restrict__ H,
    const float* __restrict__ es, const float* __restrict__ gd,
    float* __restrict__ S, float* __restrict__ Z, int E)
{
  __shared__ float zacc[HEADS];
  if (threadIdx.x < HEADS) zacc[threadIdx.x] = 0.f;
  __syncthreads();

  const int lane = threadIdx.x & 31;
  const int gw   = (blockIdx.x * blockDim.x + threadIdx.x) >> 5;
  const int nw   = (gridDim.x * blockDim.x) >> 5;

  float zp = 0.f;                      // per-lane Z partial (lanes 0..3 valid)
  for (int e = gw; e < E; e += nw) {
    const int src = eidx[e];
    const int dst = eidx[E + e];
    const float hv  = H[(size_t)src * DIM + lane];            // d = lane
    const float esv = (lane < HEADS) ? es[src * HEADS + lane] : 0.f;
    const float gdv = (lane < HEADS) ? gd[dst * HEADS + lane] : 0.f;
    zp += esv * gdv;

    float* srow = &S[(size_t)dst * DIM];
    #pragma unroll
    for (int h = 0; h < HEADS; ++h) {
      const float eh = __shfl(esv, h, 32);
      atomicAdd(&srow[h * HDIM + lane], eh * hv);
    }
  }
  if (lane < HEADS) atomicAdd(&zacc[lane], zp);               // ds_add_f32
  __syncthreads();
  if (threadIdx.x < HEADS) atomicAdd(&Z[threadIdx.x], zacc[threadIdx.x]);
}

// ---------------------------------------------------------------------------
// K4: out[n,j] = h[n,j] + (gd[n, j>>5] / Z[j>>5]) * S[n,j]   (in-place on h)
// ---------------------------------------------------------------------------
__global__ __launch_bounds__(256) void gat_finalize(
    const float* __restrict__ S, const float* __restrict__ gd,
    const float* __restrict__ Z, float* __restrict__ Out, int total)
{
  const int idx = blockIdx.x * blockDim.x + threadIdx.x;
  if (idx >= total) return;
  const int j    = idx & (DIM - 1);
  const int head = j >> 5;
  const int n    = idx >> 7;
  const float scale = gd[n * HEADS + head] / Z[head];
  Out[idx] = Out[idx] + scale * S[idx];
}

// ---------------------------------------------------------------------------
extern "C" void kernel_launch(void* const* d_in, const int* in_sizes, int n_in,
                              void* d_out, int out_size, void* d_ws, size_t ws_size,
                              hipStream_t stream) {
  const float* features = (const float*)d_in[0];   // [N,128] f32
  const int*   edges    = (const int*)d_in[1];     // [2,E]   int32
  const float* W_lin    = (const float*)d_in[2];   // [128,128]
  const float* b_lin    = (const float*)d_in[3];   // [128]
  const float* W_att    = (const float*)d_in[4];   // [4,256]
  const float* b_att    = (const float*)d_in[5];   // [4]
  float* out = (float*)d_out;                      // [N,128] (stages h, then result)

  const int nn = in_sizes[0] / DIM;                // 100000
  const int E  = in_sizes[1] / 2;                  // 500000

  // Scratch layout: S[N*128] | es[N*4] | gd[N*4] | Z[4]
  float* S  = (float*)d_ws;
  float* es = S  + (size_t)nn * DIM;
  float* gd = es + (size_t)nn * HEADS;
  float* Z  = gd + (size_t)nn * HEADS;

  hipMemsetAsync(S, 0, (size_t)nn * DIM * sizeof(float), stream);
  hipMemsetAsync(Z, 0, HEADS * sizeof(float), stream);

  const int mblocks = (nn + 15) / 16;                         // 6250
  gat_h_gemm<<<mblocks, 256, 0, stream>>>(features, W_lin, b_lin, out, nn);

  const int pblocks = (nn + 7) / 8;                           // 8 waves/block
  gat_att_proj<<<pblocks, 256, 0, stream>>>(out, W_att, b_att, es, gd, nn);

  gat_edge_scatter<<<2048, 256, 0, stream>>>(edges, out, es, gd, S, Z, E);

  const int total = nn * DIM;
  gat_finalize<<<(total + 255) / 256, 256, 0, stream>>>(S, gd, Z, out, total);
}